// TopK_18047452577798
// MI455X (gfx1250) — compile-verified
//
#include <hip/hip_runtime.h>
#include <hip/hip_bf16.h>

// Top-K (k=64) per-row masking: out = x where x is among the row's top-64, else 0.
// One workgroup per row; register-resident 8-bit radix select (4 rounds) on an
// order-preserving key transform of fp32 bits; single HBM read + single write.
// Per-round reduction is wave-synchronous (shfl-based suffix scan in wave 0)
// to minimize block-wide barriers: 1 init + 3 per round = 13 total.

#define TPB    1024              // threads per block (32 wave32 waves)
#define NWAVES (TPB / 32)
#define COLS   32768
#define EPT    (COLS / TPB)      // 32 elements per thread
#define VPT    (EPT / 4)         // 8 float4 per thread
#define KSEL   64

__global__ __launch_bounds__(TPB) void topk_mask_kernel(
    const float* __restrict__ x, float* __restrict__ out)
{
    const int row  = blockIdx.x;
    const int tid  = threadIdx.x;
    const int wv   = tid >> 5;    // wave index within workgroup
    const int lane = tid & 31;

    __shared__ unsigned hist[NWAVES * 256];  // per-wave privatized histograms (32 KB)
    __shared__ unsigned comb[256];           // combined bin counts
    __shared__ unsigned sh_prefix;
    __shared__ unsigned sh_krem;
    __shared__ unsigned sh_tie;

    const float4* __restrict__ xrow =
        reinterpret_cast<const float4*>(x + (size_t)row * COLS);
    float4* __restrict__ orow =
        reinterpret_cast<float4*>(out + (size_t)row * COLS);

    // ---- Load row (coalesced b128) and build order-preserving keys ----------
    // key = neg ? ~u : (u | 0x80000000): unsigned compare == float order; invertible.
    unsigned key[EPT];
    #pragma unroll
    for (int i = 0; i < VPT; ++i) {
        float4 v = xrow[i * TPB + tid];
        unsigned u0 = __float_as_uint(v.x);
        unsigned u1 = __float_as_uint(v.y);
        unsigned u2 = __float_as_uint(v.z);
        unsigned u3 = __float_as_uint(v.w);
        key[i * 4 + 0] = ((int)u0 < 0) ? ~u0 : (u0 | 0x80000000u);
        key[i * 4 + 1] = ((int)u1 < 0) ? ~u1 : (u1 | 0x80000000u);
        key[i * 4 + 2] = ((int)u2 < 0) ? ~u2 : (u2 | 0x80000000u);
        key[i * 4 + 3] = ((int)u3 < 0) ? ~u3 : (u3 | 0x80000000u);
    }

    // ---- One-time init: zero histograms + scalars ---------------------------
    #pragma unroll
    for (int i = 0; i < (NWAVES * 256) / TPB; ++i)
        hist[i * TPB + tid] = 0u;
    if (tid == 0) { sh_prefix = 0u; sh_krem = (unsigned)KSEL; sh_tie = 0u; }
    __syncthreads();

    unsigned prefix = 0u;

    // ---- 4 rounds of 8-bit radix select, MSB -> LSB -------------------------
    #pragma unroll
    for (int r = 0; r < 4; ++r) {
        const unsigned sh   = (unsigned)(24 - 8 * r);
        const unsigned mask = (r == 0) ? 0x00000000u
                            : (r == 1) ? 0xFF000000u
                            : (r == 2) ? 0xFFFF0000u
                            :            0xFFFFFF00u;

        // accumulate: only elements matching the selected high-bits prefix
        #pragma unroll
        for (int e = 0; e < EPT; ++e) {
            if (((key[e] ^ prefix) & mask) == 0u)
                atomicAdd(&hist[wv * 256 + ((key[e] >> sh) & 0xFFu)], 1u);
        }
        __syncthreads();

        // combine the 32 wave-private histograms AND clear them for next round
        if (tid < 256) {
            unsigned s = 0u;
            #pragma unroll
            for (int w = 0; w < NWAVES; ++w) {
                s += hist[w * 256 + tid];
                hist[w * 256 + tid] = 0u;
            }
            comb[tid] = s;
        }
        __syncthreads();

        // wave 0 only: suffix scan over 256 bins + crossing select, no barriers
        if (tid < 32) {
            unsigned kr = sh_krem;            // read before any lane writes it

            // lane owns bins [lane*8, lane*8+8); local suffix sums in registers
            unsigned c[8];
            #pragma unroll
            for (int j = 0; j < 8; ++j) c[j] = comb[lane * 8 + j];
            unsigned ls[8];
            ls[7] = c[7];
            #pragma unroll
            for (int j = 6; j >= 0; --j) ls[j] = c[j] + ls[j + 1];

            // exclusive suffix scan of lane totals across the wave (5 shfl steps)
            unsigned tot  = ls[0];
            unsigned incl = tot;
            #pragma unroll
            for (int off = 1; off < 32; off <<= 1) {
                unsigned o = __shfl_down(incl, off, 32);
                if (lane + off < 32) incl += o;
            }
            unsigned excl = incl - tot;       // sum over lanes > this lane

            // find unique bin b with S(b) >= kr > S(b+1)
            #pragma unroll
            for (int j = 0; j < 8; ++j) {
                unsigned St = excl + ls[j];
                unsigned Sn = excl + ((j < 7) ? ls[j + 1] : 0u);
                if (St >= kr && Sn < kr) {
                    sh_prefix = prefix | ((unsigned)(lane * 8 + j) << sh);
                    sh_krem   = kr - Sn;      // how many to take inside bin b
                }
            }
        }
        __syncthreads();
        prefix = sh_prefix;
    }

    const unsigned T  = prefix;    // exact key of the k-th largest element
    const unsigned kt = sh_krem;   // number of key==T elements to keep (>=1)

    // ---- Write pass: keep key > T; take kt ties at key == T -----------------
    #pragma unroll
    for (int i = 0; i < VPT; ++i) {
        float rres[4];
        #pragma unroll
        for (int j = 0; j < 4; ++j) {
            unsigned k = key[i * 4 + j];
            bool keep = (k > T);
            if (k == T) {
                unsigned p = atomicAdd(&sh_tie, 1u);
                keep = (p < kt);
            }
            // invert the key transform to recover the original float bits
            unsigned u = (k & 0x80000000u) ? (k ^ 0x80000000u) : ~k;
            rres[j] = keep ? __uint_as_float(u) : 0.0f;
        }
        float4 o;
        o.x = rres[0]; o.y = rres[1]; o.z = rres[2]; o.w = rres[3];
        orow[i * TPB + tid] = o;
    }
}

extern "C" void kernel_launch(void* const* d_in, const int* in_sizes, int n_in,
                              void* d_out, int out_size, void* d_ws, size_t ws_size,
                              hipStream_t stream) {
    const float* x = (const float*)d_in[0];
    float* out     = (float*)d_out;
    const int rows = in_sizes[0] / COLS;   // 8192 for the reference shapes
    dim3 grid((unsigned)rows), block(TPB);
    hipLaunchKernelGGL(topk_mask_kernel, grid, block, 0, stream, x, out);
}